// FlowPastBlock_36275293782435
// MI455X (gfx1250) — compile-verified
//
#include <hip/hip_runtime.h>

typedef __attribute__((ext_vector_type(2))) float v2f;
typedef __attribute__((ext_vector_type(8))) float v8f;

#define NYG 1024
#define NXG 2048
#define LSTR 66          // 64 + 2 halo columns
#define TROWS 34         // 32 + 2 halo rows
#define TELEMS (TROWS * LSTR)

namespace {
constexpr float DT      = 0.1f;
constexpr float NU      = 0.01f;
constexpr float UB      = 1.0f;
constexpr float INVDIAG = -3.0f / 8.0f;   // 1 / (-8/(3*dx*dx)), dx=1
constexpr float THIRD   = 1.0f / 3.0f;
constexpr float W112    = 1.0f / 12.0f;
constexpr float W412    = 4.0f / 12.0f;
}

#if defined(__AMDGCN__) && __has_builtin(__builtin_amdgcn_wmma_f32_16x16x4_f32)
#define HAVE_WMMA_F32 1
#else
#define HAVE_WMMA_F32 0
#endif

// ---------------- boundary-condition fetch (matches reference _bc_* exactly) --
// MODE 0: bc_u   MODE 1: bc_v   MODE 2: bc_p
template <int MODE>
__device__ __forceinline__ float bc_fetch(const float* __restrict__ f, int gy, int gx) {
  if (MODE == 0) {                       // u: top/bottom/left = UB, right zero-grad
    if (gy < 0 || gy >= NYG) return UB;
    if (gx < 0) return UB;
    if (gx >= NXG) return f[gy * NXG + (NXG - 1)];
    return f[gy * NXG + gx];
  } else if (MODE == 1) {                // v: zero halo, right zero-grad
    if (gy < 0 || gy >= NYG) return 0.0f;
    if (gx < 0) return 0.0f;
    if (gx >= NXG) return f[gy * NXG + (NXG - 1)];
    return f[gy * NXG + gx];
  } else {                               // p: zero-grad except p=0 at outlet
    if (gx >= NXG) return 0.0f;
    int cy = gy < 0 ? 0 : (gy >= NYG ? NYG - 1 : gy);
    int cx = gx < 0 ? 0 : gx;
    return f[cy * NXG + cx];
  }
}

template <int MODE>
__device__ __forceinline__ void load_tile(float* __restrict__ lds,
                                          const float* __restrict__ f,
                                          int by, int bx) {
  for (int idx = threadIdx.x; idx < TELEMS; idx += 256) {
    int ly = idx / LSTR;
    int lx = idx - ly * LSTR;
    lds[idx] = bc_fetch<MODE>(f, by - 1 + ly, bx - 1 + lx);
  }
}

// tile of t = p - prolong(X1) with bc_p applied to t (for the MG smoothing step)
__device__ __forceinline__ void load_tile_t(float* __restrict__ lds,
                                            const float* __restrict__ p,
                                            const float* __restrict__ X1,
                                            int by, int bx) {
  for (int idx = threadIdx.x; idx < TELEMS; idx += 256) {
    int ly = idx / LSTR;
    int lx = idx - ly * LSTR;
    int gy = by - 1 + ly, gx = bx - 1 + lx;
    float val;
    if (gx >= NXG) {
      val = 0.0f;
    } else {
      int cy = gy < 0 ? 0 : (gy >= NYG ? NYG - 1 : gy);
      int cx = gx < 0 ? 0 : gx;
      val = p[cy * NXG + cx] - X1[(cy >> 1) * (NXG >> 1) + (cx >> 1)];
    }
    lds[idx] = val;
  }
}

// ---------------- separable 3x3 conv over a 16x16 tile ------------------------
// y-tap via VALU from LDS, x-tap as banded 16x16 matmul via V_WMMA_F32_16X16X4_F32
__device__ __forceinline__ float yfil(const float* __restrict__ t, int y, int x,
                                      float wy0, float wy1, float wy2) {
  return wy0 * t[(y - 1) * LSTR + x] + wy1 * t[y * LSTR + x] + wy2 * t[(y + 1) * LSTR + x];
}

__device__ __forceinline__ float sxval(int k, int n, float wx0, float wx1, float wx2) {
  float s = 0.0f;
  s = (n == k - 1) ? wx0 : s;
  s = (n == k)     ? wx1 : s;
  s = (n == k + 1) ? wx2 : s;
  return s;
}

// Result in WMMA C/D layout: VGPR v holds row m = v + 8*(lane>=16), col n = lane&15.
__device__ __forceinline__ v8f conv_sep(const float* __restrict__ t, int oy, int ox,
                                        float wy0, float wy1, float wy2,
                                        float wx0, float wx1, float wx2) {
  const int lane = threadIdx.x & 31;
  const int h = lane >> 4;
  const int r = lane & 15;
  v8f acc = {0.f, 0.f, 0.f, 0.f, 0.f, 0.f, 0.f, 0.f};
#if HAVE_WMMA_F32
#pragma unroll
  for (int j = 0; j < 4; ++j) {
    const int k0 = 4 * j + 2 * h;
    v2f aa, bb;
    aa.x = yfil(t, oy + r, ox + k0,     wy0, wy1, wy2);   // A: lane=row, K across VGPRs
    aa.y = yfil(t, oy + r, ox + k0 + 1, wy0, wy1, wy2);
    bb.x = sxval(k0,     r, wx0, wx1, wx2);               // B: banded x-stencil matrix
    bb.y = sxval(k0 + 1, r, wx0, wx1, wx2);
    acc = __builtin_amdgcn_wmma_f32_16x16x4_f32(false, aa, false, bb,
                                                (short)0, acc, false, false);
  }
  // halo corrections: output col 0 needs x=-1, col 15 needs x=16
  if (r == 0) {
#pragma unroll
    for (int v = 0; v < 8; ++v)
      acc[v] += wx0 * yfil(t, oy + v + 8 * h, ox - 1, wy0, wy1, wy2);
  } else if (r == 15) {
#pragma unroll
    for (int v = 0; v < 8; ++v)
      acc[v] += wx2 * yfil(t, oy + v + 8 * h, ox + 16, wy0, wy1, wy2);
  }
#else
#pragma unroll
  for (int v = 0; v < 8; ++v) {
    const int m = v + 8 * h;
    float t0 = yfil(t, oy + m, ox + r - 1, wy0, wy1, wy2);
    float t1 = yfil(t, oy + m, ox + r,     wy0, wy1, wy2);
    float t2 = yfil(t, oy + m, ox + r + 1, wy0, wy1, wy2);
    acc[v] = wx0 * t0 + wx1 * t1 + wx2 * t2;
  }
#endif
  return acc;
}

__device__ __forceinline__ v8f conv_xadv(const float* t, int oy, int ox) {
  return conv_sep(t, oy, ox, 1.f, 4.f, 1.f, -W112, 0.f, W112);
}
__device__ __forceinline__ v8f conv_yadv(const float* t, int oy, int ox) {
  return conv_sep(t, oy, ox, -1.f, 0.f, 1.f, W112, W412, W112);
}
__device__ __forceinline__ v8f conv_sum9(const float* t, int oy, int ox) {
  return conv_sep(t, oy, ox, 1.f, 1.f, 1.f, 1.f, 1.f, 1.f);
}

__device__ __forceinline__ v8f center_tile(const float* __restrict__ t, int oy, int ox) {
  const int lane = threadIdx.x & 31;
  const int h = lane >> 4, n = lane & 15;
  v8f c;
#pragma unroll
  for (int v = 0; v < 8; ++v) c[v] = t[(oy + v + 8 * h) * LSTR + ox + n];
  return c;
}

__device__ __forceinline__ v8f gather_g(const float* __restrict__ f, int gy0, int gx0) {
  const int lane = threadIdx.x & 31;
  const int h = lane >> 4, n = lane & 15;
  v8f c;
#pragma unroll
  for (int v = 0; v < 8; ++v) c[v] = f[(gy0 + v + 8 * h) * NXG + gx0 + n];
  return c;
}

__device__ __forceinline__ void store_g(float* __restrict__ f, int gy0, int gx0, v8f val) {
  const int lane = threadIdx.x & 31;
  const int h = lane >> 4, n = lane & 15;
#pragma unroll
  for (int v = 0; v < 8; ++v) f[(gy0 + v + 8 * h) * NXG + gx0 + n] = val[v];
}

#define TILE_SETUP()                                             \
  const int wave = threadIdx.x >> 5;                             \
  const int wtr = wave >> 2, wtc = wave & 3;                     \
  const int oy = 1 + 16 * wtr, ox = 1 + 16 * wtc;                \
  const int gy0 = blockIdx.y * 32 + 16 * wtr;                    \
  const int gx0 = blockIdx.x * 64 + 16 * wtc;

// ---------------- fine-grid kernels -----------------------------------------
__global__ __launch_bounds__(256) void k_predictor(
    const float* __restrict__ u, const float* __restrict__ v,
    const float* __restrict__ p, const float* __restrict__ sigma,
    float* __restrict__ bu, float* __restrict__ bv) {
  __shared__ float sU[TELEMS], sV[TELEMS], sP[TELEMS];
  const int byl = blockIdx.y * 32, bxl = blockIdx.x * 64;
  load_tile<0>(sU, u, byl, bxl);
  load_tile<1>(sV, v, byl, bxl);
  load_tile<2>(sP, p, byl, bxl);
  __syncthreads();
  TILE_SETUP();
  v8f uc = center_tile(sU, oy, ox);
  v8f vc = center_tile(sV, oy, ox);
  v8f Gx = conv_xadv(sP, oy, ox);
  v8f Gy = conv_yadv(sP, oy, ox);
  v8f ADxu = conv_xadv(sU, oy, ox);
  v8f ADyu = conv_yadv(sU, oy, ox);
  v8f ADxv = conv_xadv(sV, oy, ox);
  v8f ADyv = conv_yadv(sV, oy, ox);
  v8f Su = conv_sum9(sU, oy, ox);
  v8f Sv = conv_sum9(sV, oy, ox);
  v8f sg = gather_g(sigma, gy0, gx0);
  v8f obu, obv;
#pragma unroll
  for (int i = 0; i < 8; ++i) {
    float AD2u = (Su[i] - 9.0f * uc[i]) * THIRD;
    float AD2v = (Sv[i] - 9.0f * vc[i]) * THIRD;
    float bu_ = uc[i] + 0.5f * (NU * AD2u * DT - uc[i] * ADxu[i] * DT - vc[i] * ADyu[i] * DT) - Gx[i] * DT;
    float bv_ = vc[i] + 0.5f * (NU * AD2v * DT - uc[i] * ADxv[i] * DT - vc[i] * ADyv[i] * DT) - Gy[i] * DT;
    float d = 1.0f / (1.0f + DT * sg[i]);
    obu[i] = bu_ * d;
    obv[i] = bv_ * d;
  }
  store_g(bu, gy0, gx0, obu);
  store_g(bv, gy0, gx0, obv);
}

__global__ __launch_bounds__(256) void k_corrector(
    const float* __restrict__ u, const float* __restrict__ v,
    const float* __restrict__ p, const float* __restrict__ sigma,
    const float* __restrict__ bu, const float* __restrict__ bv,
    float* __restrict__ us, float* __restrict__ vs) {
  __shared__ float sBU[TELEMS], sBV[TELEMS], sP[TELEMS];
  const int byl = blockIdx.y * 32, bxl = blockIdx.x * 64;
  load_tile<0>(sBU, bu, byl, bxl);
  load_tile<1>(sBV, bv, byl, bxl);
  load_tile<2>(sP, p, byl, bxl);
  __syncthreads();
  TILE_SETUP();
  v8f buc = center_tile(sBU, oy, ox);
  v8f bvc = center_tile(sBV, oy, ox);
  v8f Gx = conv_xadv(sP, oy, ox);
  v8f Gy = conv_yadv(sP, oy, ox);
  v8f ADxu = conv_xadv(sBU, oy, ox);
  v8f ADyu = conv_yadv(sBU, oy, ox);
  v8f ADxv = conv_xadv(sBV, oy, ox);
  v8f ADyv = conv_yadv(sBV, oy, ox);
  v8f Su = conv_sum9(sBU, oy, ox);
  v8f Sv = conv_sum9(sBV, oy, ox);
  v8f ug = gather_g(u, gy0, gx0);
  v8f vg = gather_g(v, gy0, gx0);
  v8f sg = gather_g(sigma, gy0, gx0);
  v8f ou, ov;
#pragma unroll
  for (int i = 0; i < 8; ++i) {
    float AD2u = (Su[i] - 9.0f * buc[i]) * THIRD;
    float AD2v = (Sv[i] - 9.0f * bvc[i]) * THIRD;
    float un = ug[i] + NU * AD2u * DT - buc[i] * ADxu[i] * DT - bvc[i] * ADyu[i] * DT - Gx[i] * DT;
    float vn = vg[i] + NU * AD2v * DT - buc[i] * ADxv[i] * DT - bvc[i] * ADyv[i] * DT - Gy[i] * DT;
    float d = 1.0f / (1.0f + DT * sg[i]);
    ou[i] = un * d;
    ov[i] = vn * d;
  }
  store_g(us, gy0, gx0, ou);
  store_g(vs, gy0, gx0, ov);
}

__global__ __launch_bounds__(256) void k_divcopy(
    const float* __restrict__ us, const float* __restrict__ vs,
    const float* __restrict__ p_in, float* __restrict__ b, float* __restrict__ p_cur) {
  __shared__ float sU[TELEMS], sV[TELEMS];
  const int byl = blockIdx.y * 32, bxl = blockIdx.x * 64;
  load_tile<0>(sU, us, byl, bxl);
  load_tile<1>(sV, vs, byl, bxl);
  __syncthreads();
  TILE_SETUP();
  v8f dx = conv_xadv(sU, oy, ox);
  v8f dy = conv_yadv(sV, oy, ox);
  v8f ob;
#pragma unroll
  for (int i = 0; i < 8; ++i) ob[i] = -(dx[i] + dy[i]) * (1.0f / DT);
  store_g(b, gy0, gx0, ob);
  v8f pg = gather_g(p_in, gy0, gx0);
  store_g(p_cur, gy0, gx0, pg);
}

__global__ __launch_bounds__(256) void k_residual(
    const float* __restrict__ p, const float* __restrict__ b, float* __restrict__ r0) {
  __shared__ float sP[TELEMS];
  const int byl = blockIdx.y * 32, bxl = blockIdx.x * 64;
  load_tile<2>(sP, p, byl, bxl);
  __syncthreads();
  TILE_SETUP();
  v8f pcen = center_tile(sP, oy, ox);
  v8f S = conv_sum9(sP, oy, ox);
  v8f bg = gather_g(b, gy0, gx0);
  v8f orr;
#pragma unroll
  for (int i = 0; i < 8; ++i) orr[i] = (S[i] - 9.0f * pcen[i]) * THIRD - bg[i];
  store_g(r0, gy0, gx0, orr);
}

__global__ __launch_bounds__(256) void k_update_p(
    const float* __restrict__ p, const float* __restrict__ X1,
    const float* __restrict__ b, float* __restrict__ p_out) {
  __shared__ float sT[TELEMS];
  const int byl = blockIdx.y * 32, bxl = blockIdx.x * 64;
  load_tile_t(sT, p, X1, byl, bxl);
  __syncthreads();
  TILE_SETUP();
  v8f tcen = center_tile(sT, oy, ox);
  v8f S = conv_sum9(sT, oy, ox);
  v8f bg = gather_g(b, gy0, gx0);
  v8f op;
#pragma unroll
  for (int i = 0; i < 8; ++i) {
    float conv = (S[i] - 9.0f * tcen[i]) * THIRD;
    op[i] = tcen[i] - conv * INVDIAG + bg[i] * INVDIAG;
  }
  store_g(p_out, gy0, gx0, op);
}

__global__ __launch_bounds__(256) void k_final(
    const float* __restrict__ p, const float* __restrict__ sigma,
    float* __restrict__ out_u, float* __restrict__ out_v, float* __restrict__ out_p) {
  __shared__ float sP[TELEMS];
  const int byl = blockIdx.y * 32, bxl = blockIdx.x * 64;
  load_tile<2>(sP, p, byl, bxl);
  __syncthreads();
  TILE_SETUP();
  v8f Gx = conv_xadv(sP, oy, ox);
  v8f Gy = conv_yadv(sP, oy, ox);
  v8f pcen = center_tile(sP, oy, ox);
  v8f ug = gather_g(out_u, gy0, gx0);
  v8f vg = gather_g(out_v, gy0, gx0);
  v8f sg = gather_g(sigma, gy0, gx0);
  v8f ou, ov;
#pragma unroll
  for (int i = 0; i < 8; ++i) {
    float d = 1.0f / (1.0f + DT * sg[i]);
    ou[i] = (ug[i] - Gx[i] * DT) * d;
    ov[i] = (vg[i] - Gy[i] * DT) * d;
  }
  store_g(out_u, gy0, gx0, ou);
  store_g(out_v, gy0, gx0, ov);
  store_g(out_p, gy0, gx0, pcen);
}

// ---------------- coarse multigrid kernels (tiny, VALU) ----------------------
__global__ void k_restrict(const float* __restrict__ src, float* __restrict__ dst,
                           int dh, int dw) {
  int i = blockIdx.x * 256 + threadIdx.x;
  if (i >= dh * dw) return;
  int y = i / dw, x = i - y * dw;
  int sw = dw * 2;
  dst[i] = 0.25f * (src[(2 * y) * sw + 2 * x] + src[(2 * y) * sw + 2 * x + 1] +
                    src[(2 * y + 1) * sw + 2 * x] + src[(2 * y + 1) * sw + 2 * x + 1]);
}

__global__ void k_coarsest(const float* __restrict__ r10, float* __restrict__ X10) {
  if (threadIdx.x < 2) X10[threadIdx.x] = r10[threadIdx.x] * INVDIAG;
}

__device__ __forceinline__ float upz(const float* __restrict__ Xc, int y, int x,
                                     int h, int w, int cw) {
  if (y < 0 || y >= h || x < 0 || x >= w) return 0.0f;
  return Xc[(y >> 1) * cw + (x >> 1)];
}

__global__ void k_mg_up(const float* __restrict__ Xc, const float* __restrict__ r,
                        float* __restrict__ Xout, int h, int w) {
  int i = blockIdx.x * 256 + threadIdx.x;
  if (i >= h * w) return;
  int y = i / w, x = i - y * w;
  int cw = w >> 1;
  float c = upz(Xc, y, x, h, w, cw);
  float s = upz(Xc, y - 1, x - 1, h, w, cw) + upz(Xc, y - 1, x, h, w, cw) +
            upz(Xc, y - 1, x + 1, h, w, cw) + upz(Xc, y, x - 1, h, w, cw) +
            upz(Xc, y, x + 1, h, w, cw) + upz(Xc, y + 1, x - 1, h, w, cw) +
            upz(Xc, y + 1, x, h, w, cw) + upz(Xc, y + 1, x + 1, h, w, cw);
  float conv = (s - 8.0f * c) * THIRD;   // 3x3 [-8 center] stencil, zero halo
  Xout[i] = c - conv * INVDIAG + r[i] * INVDIAG;
}

// ---------------- host orchestration -----------------------------------------
extern "C" void kernel_launch(void* const* d_in, const int* in_sizes, int n_in,
                              void* d_out, int out_size, void* d_ws, size_t ws_size,
                              hipStream_t stream) {
  (void)in_sizes; (void)n_in; (void)out_size; (void)ws_size;
  const float* u     = (const float*)d_in[0];
  const float* v     = (const float*)d_in[1];
  const float* p     = (const float*)d_in[2];
  const float* sigma = (const float*)d_in[3];
  // d_in[4..8] are the stencil weights; they are compile-time constants here.

  const size_t N0 = (size_t)NYG * NXG;
  float* out_u = (float*)d_out;
  float* out_v = out_u + N0;
  float* out_p = out_u + 2 * N0;

  float* ws = (float*)d_ws;
  float* bu = ws;            // also hosts the X pyramid (bu dead after corrector)
  float* bv = ws + N0;       // also hosts r level 0 (bv dead after corrector)
  float* bb = ws + 2 * N0;
  float* pA = ws + 3 * N0;
  float* pB = ws + 4 * N0;

  size_t lvl_n[11];
  for (int i = 0; i < 11; ++i) lvl_n[i] = (size_t)(NYG >> i) * (size_t)(NXG >> i);

  float* rlev[11];
  rlev[0] = bv;
  {
    size_t off = 5 * N0;
    for (int i = 1; i < 11; ++i) { rlev[i] = ws + off; off += lvl_n[i]; }
  }
  float* xlev[11];
  {
    size_t off = 0;
    for (int i = 1; i < 11; ++i) { xlev[i] = bu + off; off += lvl_n[i]; }
  }

  dim3 grid(NXG / 64, NYG / 32);
  dim3 block(256);

  k_predictor<<<grid, block, 0, stream>>>(u, v, p, sigma, bu, bv);
  k_corrector<<<grid, block, 0, stream>>>(u, v, p, sigma, bu, bv, out_u, out_v);
  k_divcopy<<<grid, block, 0, stream>>>(out_u, out_v, p, bb, pA);

  float* pc = pA;
  float* pn = pB;
  for (int it = 0; it < 5; ++it) {
    k_residual<<<grid, block, 0, stream>>>(pc, bb, rlev[0]);
    for (int i = 0; i < 10; ++i) {
      int dh = NYG >> (i + 1), dw = NXG >> (i + 1);
      int n = dh * dw;
      k_restrict<<<(n + 255) / 256, 256, 0, stream>>>(rlev[i], rlev[i + 1], dh, dw);
    }
    k_coarsest<<<1, 32, 0, stream>>>(rlev[10], xlev[10]);
    for (int i = 9; i >= 1; --i) {
      int h = NYG >> i, w = NXG >> i;
      int n = h * w;
      k_mg_up<<<(n + 255) / 256, 256, 0, stream>>>(xlev[i + 1], rlev[i], xlev[i], h, w);
    }
    k_update_p<<<grid, block, 0, stream>>>(pc, xlev[1], bb, pn);
    float* t = pc; pc = pn; pn = t;
  }

  k_final<<<grid, block, 0, stream>>>(pc, sigma, out_u, out_v, out_p);
}